// TAGWithJK_76776835383358
// MI455X (gfx1250) — compile-verified
//
#include <hip/hip_runtime.h>

#define NN 100000
#define EE 3200000
#define GG 64
#define FIN 128
#define HH 8

typedef __attribute__((ext_vector_type(2))) float v2f;
typedef __attribute__((ext_vector_type(8))) float v8f;

// -------------------- utility --------------------
__global__ void zero_f32(float* __restrict__ p, int n) {
  int i = blockIdx.x * blockDim.x + threadIdx.x;
  if (i < n) p[i] = 0.f;
}

// -------------------- GCN normalization --------------------
__global__ void deg_acc(const int* __restrict__ dst, const float* __restrict__ w,
                        float* __restrict__ deg) {
  int e = blockIdx.x * blockDim.x + threadIdx.x;
  if (e < EE) atomicAdd(&deg[dst[e]], w[e]);
}

__global__ void dis_k(const float* __restrict__ deg, float* __restrict__ dis) {
  int n = blockIdx.x * blockDim.x + threadIdx.x;
  if (n < NN) { float d = deg[n]; dis[n] = (d > 0.f) ? rsqrtf(d) : 0.f; }
}

__global__ void norm_k(const int* __restrict__ src, const int* __restrict__ dst,
                       const float* __restrict__ w, const float* __restrict__ dis,
                       float* __restrict__ nrm) {
  int e = blockIdx.x * blockDim.x + threadIdx.x;
  if (e < EE) nrm[e] = dis[src[e]] * w[e] * dis[dst[e]];
}

// -------------------- WMMA projection GEMM --------------------
// Z[N x 32] = A[N x KDIM] @ Wcat[KDIM x 32],
// Wcat[k][n] = W[(n>>3)*KDIM + k][n&7]  (W is the reference ((K+1)*KDIM, 8) matrix)
// One wave computes a 16-row x 32-col tile with V_WMMA_F32_16X16X4_F32.
// NN % 16 == 0, so per-wave guard is wave-uniform and EXEC stays all-ones.
template <int KDIM>
__global__ void wmma_proj(const float* __restrict__ A, const float* __restrict__ W,
                          float* __restrict__ Z, int nrows) {
  int wave = threadIdx.x >> 5;
  int lane = threadIdx.x & 31;
  int rowBase = (blockIdx.x * (blockDim.x >> 5) + wave) * 16;
  if (rowBase >= nrows) return;  // uniform across the wave

  int m = lane & 15;
  int khalf = (lane >> 4) * 2;  // lanes 0-15 hold K={0,1}, lanes 16-31 hold K={2,3}
  const float* arow = A + (size_t)(rowBase + m) * KDIM;

  // Per-lane B base pointers: column n0 = m (tile 0), n1 = 16+m (tile 1).
  const float* wb0 = W + (size_t)((m >> 3) * KDIM) * 8 + (m & 7);
  const float* wb1 = W + (size_t)((2 + (m >> 3)) * KDIM) * 8 + (m & 7);

  v8f c0 = {};
  v8f c1 = {};
#pragma unroll 4
  for (int k = 0; k < KDIM; k += 4) {
    int kk = k + khalf;
    v2f a, b0, b1;
    a.x = arow[kk + 0];
    a.y = arow[kk + 1];
    b0.x = wb0[(size_t)(kk + 0) * 8];
    b0.y = wb0[(size_t)(kk + 1) * 8];
    b1.x = wb1[(size_t)(kk + 0) * 8];
    b1.y = wb1[(size_t)(kk + 1) * 8];
    c0 = __builtin_amdgcn_wmma_f32_16x16x4_f32(false, a, false, b0, (short)0, c0,
                                               false, false);
    c1 = __builtin_amdgcn_wmma_f32_16x16x4_f32(false, a, false, b1, (short)0, c1,
                                               false, false);
  }

  // C/D layout: VGPR v, lanes 0-15 -> row v, col lane; lanes 16-31 -> row v+8, col lane-16
  int mrow = rowBase + (lane >> 4) * 8;
  int col = lane & 15;
#pragma unroll
  for (int v = 0; v < 8; ++v) {
    Z[(size_t)(mrow + v) * 32 + col] = c0[v];
    Z[(size_t)(mrow + v) * 32 + 16 + col] = c1[v];
  }
}

// -------------------- Horner propagation --------------------
__global__ void init_a(const float* __restrict__ z, float* __restrict__ a) {
  int i = blockIdx.x * blockDim.x + threadIdx.x;  // N*8
  if (i < NN * 8) {
    int n = i >> 3, j = i & 7;
    a[i] = z[(size_t)n * 32 + 24 + j];
  }
}

__global__ void edge_scatter(const int* __restrict__ src, const int* __restrict__ dst,
                             const float* __restrict__ nrm, const float* __restrict__ a,
                             float* __restrict__ out) {
  int e = blockIdx.x * blockDim.x + threadIdx.x;
  if (e >= EE) return;
  int s = src[e], d = dst[e];
  float w = nrm[e];
  const float4* ap = (const float4*)(a + (size_t)s * 8);
  float4 lo = ap[0], hi = ap[1];
  float* op = out + (size_t)d * 8;
  atomicAdd(op + 0, lo.x * w);
  atomicAdd(op + 1, lo.y * w);
  atomicAdd(op + 2, lo.z * w);
  atomicAdd(op + 3, lo.w * w);
  atomicAdd(op + 4, hi.x * w);
  atomicAdd(op + 5, hi.y * w);
  atomicAdd(op + 6, hi.z * w);
  atomicAdd(op + 7, hi.w * w);
}

// aout = t + z[:, 8*hop : 8*hop+8]; if final: aout = relu(aout + bias)
__global__ void combine(const float* __restrict__ t, const float* __restrict__ z,
                        int hop, const float* __restrict__ bias, int do_relu,
                        float* __restrict__ aout) {
  int i = blockIdx.x * blockDim.x + threadIdx.x;  // N*8
  if (i >= NN * 8) return;
  int n = i >> 3, j = i & 7;
  float v = t[i] + z[(size_t)n * 32 + hop * 8 + j];
  if (do_relu) {
    v += bias[j];
    v = fmaxf(v, 0.f);
  }
  aout[i] = v;
}

// -------------------- pooling + FC --------------------
__global__ void pool_acc(const float* __restrict__ h1, const float* __restrict__ h2,
                         const float* __restrict__ h3, const int* __restrict__ batch,
                         unsigned int* __restrict__ gmax, float* __restrict__ gsum,
                         float* __restrict__ cnt) {
  int i = blockIdx.x * blockDim.x + threadIdx.x;  // N*24
  if (i >= NN * 24) return;
  int n = i / 24, j = i - n * 24;
  float v;
  if (j < 8)
    v = h1[(size_t)n * 8 + j];
  else if (j < 16)
    v = h2[(size_t)n * 8 + (j - 8)];
  else
    v = h3[(size_t)n * 8 + (j - 16)];
  int g = batch[n];
  // values are post-ReLU (>= 0): uint-bit compare is order-preserving
  atomicMax(&gmax[(size_t)g * 24 + j], __float_as_uint(v));
  atomicAdd(&gsum[(size_t)g * 24 + j], v);
  if (j == 0) atomicAdd(&cnt[g], 1.0f);
}

__global__ void final_fc(const unsigned int* __restrict__ gmax,
                         const float* __restrict__ gsum, const float* __restrict__ cnt,
                         const float* __restrict__ fcW, const float* __restrict__ fcb,
                         float* __restrict__ out) {
  int i = blockIdx.x * blockDim.x + threadIdx.x;  // G*2
  if (i >= GG * 2) return;
  int g = i >> 1, c = i & 1;
  float inv = 1.0f / fmaxf(cnt[g], 1.0f);
  float acc = fcb[c];
#pragma unroll
  for (int j = 0; j < 24; ++j)
    acc += __uint_as_float(gmax[(size_t)g * 24 + j]) * fcW[j * 2 + c];
#pragma unroll
  for (int j = 0; j < 24; ++j)
    acc += (gsum[(size_t)g * 24 + j] * inv) * fcW[(24 + j) * 2 + c];
  out[(size_t)g * 2 + c] = acc;
}

// -------------------- host side --------------------
static inline int gridFor(long long n, int b) { return (int)((n + b - 1) / b); }

extern "C" void kernel_launch(void* const* d_in, const int* in_sizes, int n_in,
                              void* d_out, int out_size, void* d_ws, size_t ws_size,
                              hipStream_t stream) {
  const float* x = (const float*)d_in[0];
  const int* ei = (const int*)d_in[1];
  const int* src = ei;
  const int* dst = ei + EE;
  const int* batch = (const int*)d_in[2];
  const float* ew = (const float*)d_in[3];
  const float* W1 = (const float*)d_in[4];
  const float* b1 = (const float*)d_in[5];
  const float* W2 = (const float*)d_in[6];
  const float* b2 = (const float*)d_in[7];
  const float* W3 = (const float*)d_in[8];
  const float* b3 = (const float*)d_in[9];
  const float* fcW = (const float*)d_in[10];
  const float* fcb = (const float*)d_in[11];
  float* out = (float*)d_out;

  float* ws = (float*)d_ws;
  float* z = ws;                 // N*32
  float* bufA = z + (size_t)NN * 32;    // N*8
  float* bufB = bufA + (size_t)NN * 8;  // N*8
  float* h1 = bufB + (size_t)NN * 8;    // N*8
  float* h2 = h1 + (size_t)NN * 8;      // N*8
  float* h3 = h2 + (size_t)NN * 8;      // N*8
  float* deg = h3 + (size_t)NN * 8;     // N
  float* dis = deg + NN;                // N
  float* nrm = dis + NN;                // E
  float* gmaxf = nrm + EE;              // G*24 (uint bits)
  float* gsum = gmaxf + GG * 24;        // G*24
  float* cnt = gsum + GG * 24;          // G

  const int B = 256;

  // ---- normalization ----
  zero_f32<<<gridFor(NN, B), B, 0, stream>>>(deg, NN);
  deg_acc<<<gridFor(EE, B), B, 0, stream>>>(dst, ew, deg);
  dis_k<<<gridFor(NN, B), B, 0, stream>>>(deg, dis);
  norm_k<<<gridFor(EE, B), B, 0, stream>>>(src, dst, ew, dis, nrm);

  const int wavesPerBlk = B / 32;
  const int nWaves = NN / 16;  // 100000 % 16 == 0
  const int gemmGrid = gridFor(nWaves, wavesPerBlk);
  const int ne8 = NN * 8;

  const float* layerW[3] = {W1, W2, W3};
  const float* layerB[3] = {b1, b2, b3};
  float* layerH[3] = {h1, h2, h3};
  const float* hin = x;

  for (int layer = 0; layer < 3; ++layer) {
    // projection GEMM: z = hin @ Wcat  (WMMA f32 16x16x4)
    if (layer == 0)
      wmma_proj<FIN><<<gemmGrid, B, 0, stream>>>(hin, layerW[layer], z, NN);
    else
      wmma_proj<HH><<<gemmGrid, B, 0, stream>>>(hin, layerW[layer], z, NN);

    // Horner: a = z3; for hop 2,1,0: a = T a + z_hop  (final: +bias, relu)
    init_a<<<gridFor(ne8, B), B, 0, stream>>>(z, bufA);
    for (int hop = 2; hop >= 0; --hop) {
      zero_f32<<<gridFor(ne8, B), B, 0, stream>>>(bufB, ne8);
      edge_scatter<<<gridFor(EE, B), B, 0, stream>>>(src, dst, nrm, bufA, bufB);
      if (hop == 0)
        combine<<<gridFor(ne8, B), B, 0, stream>>>(bufB, z, 0, layerB[layer], 1,
                                                   layerH[layer]);
      else
        combine<<<gridFor(ne8, B), B, 0, stream>>>(bufB, z, hop, layerB[layer], 0,
                                                   bufA);
    }
    hin = layerH[layer];
  }

  // ---- pooling + FC ----
  zero_f32<<<gridFor(GG * 24 * 2 + GG, B), B, 0, stream>>>(gmaxf, GG * 24 * 2 + GG);
  pool_acc<<<gridFor((long long)NN * 24, B), B, 0, stream>>>(
      h1, h2, h3, batch, (unsigned int*)gmaxf, gsum, cnt);
  final_fc<<<1, 128, 0, stream>>>((const unsigned int*)gmaxf, gsum, cnt, fcW, fcb, out);
}